// DifferentiableWeightedRadialFrequencyLoss_42399917146943
// MI455X (gfx1250) — compile-verified
//
#include <hip/hip_runtime.h>

// DifferentiableWeightedRadialFrequencyLoss on MI455X (gfx1250)
// FFT2 as DFT-matrix GEMMs on the f32 WMMA pipe + Hermitian symmetry.
// Stage 2 stages shared B tiles in LDS via global_load_async_to_lds_b128
// (ASYNCcnt), double-buffered.

typedef __attribute__((ext_vector_type(2))) float v2f;
typedef __attribute__((ext_vector_type(8))) float v8f;

#define HW    512
#define NB    16
#define NIMG  96            // N*C = 32*3
#define CHUNK 8
#define TROWS 272           // T columns needed: 0..271
#define NCHW  (32*3*512*512)
#define KC    64            // stage-2 K chunk staged in LDS
#define LDSTR 68            // padded row stride (dwords): 16B-aligned, bank-safe

__device__ __forceinline__ v8f wmma_f32(v2f a, v2f b, v8f c) {
    return __builtin_amdgcn_wmma_f32_16x16x4_f32(
        false, a, false, b, (short)0, c, false, false);
}

__device__ __forceinline__ void async_b128(float* lds_dst, const float* gsrc) {
    unsigned lds = (unsigned)(unsigned long long)lds_dst;   // addr[31:0] = LDS offset
    unsigned long long ga = (unsigned long long)gsrc;
    asm volatile("global_load_async_to_lds_b128 %0, %1, off"
                 :: "v"(lds), "v"(ga) : "memory");
}
__device__ __forceinline__ void wait_async(int n) {
    if (n == 0) asm volatile("s_wait_asynccnt 0x0" ::: "memory");
    else        asm volatile("s_wait_asynccnt 0x4" ::: "memory");
}

// ---------------------------------------------------------------------------
// Kernel 0: W = exp(-2*pi*i*j*k/512)/sqrt(512) (symmetric), zero P
// ---------------------------------------------------------------------------
__global__ void rfl_init(float* __restrict__ Wr, float* __restrict__ Wi,
                         float* __restrict__ P) {
    int idx = blockIdx.x * blockDim.x + threadIdx.x;
    int j = idx >> 9;
    int k = idx & 511;
    int t = (j * k) & 511;
    float ang = -6.2831853071795864769f * (float)t * (1.0f / 512.0f);
    const float s = 0.04419417382415922f;               // 1/sqrt(512)
    Wr[idx] = s * cosf(ang);
    Wi[idx] = s * sinf(ang);
    P[idx]  = 0.0f;
}

// ---------------------------------------------------------------------------
// Stage 1: T = E * W, E = pred - gt fused into the A fragment.
// Only T columns 0..271 produced; output stored TRANSPOSED (Tt[n][m]).
// Grid: (17, 8, chunk), block 128 (4 waves; wave w -> m0 = by*64 + w*16).
// ---------------------------------------------------------------------------
__global__ void rfl_stage1(const float* __restrict__ pred,
                           const float* __restrict__ gt,
                           const float* __restrict__ Wr,
                           const float* __restrict__ Wi,
                           float* __restrict__ TrT,
                           float* __restrict__ TiT,
                           int img0) {
    const int wave = threadIdx.x >> 5;
    const int lane = threadIdx.x & 31;
    const int half = lane >> 4;
    const int r    = lane & 15;
    const int m0 = blockIdx.y * 64 + wave * 16;
    const int n0 = blockIdx.x * 16;

    const size_t ib = (size_t)(img0 + blockIdx.z) * HW * HW;
    const float* p = pred + ib;
    const float* g = gt   + ib;
    float* trt = TrT + (size_t)blockIdx.z * TROWS * HW;
    float* tit = TiT + (size_t)blockIdx.z * TROWS * HW;

    const int arow = m0 + r;
    const int ncol = n0 + r;

    v8f accR = {};
    v8f accI = {};
    for (int k0 = 0; k0 < HW; k0 += 4) {
        const int ka = k0 + 2 * half;
        v2f a  = *(const v2f*)(p + arow * HW + ka);
        v2f gg = *(const v2f*)(g + arow * HW + ka);
        a.x -= gg.x;
        a.y -= gg.y;
        v2f br = *(const v2f*)(Wr + ncol * HW + ka);   // W symmetric
        v2f bi = *(const v2f*)(Wi + ncol * HW + ka);
        accR = wmma_f32(a, br, accR);
        accI = wmma_f32(a, bi, accI);
    }
    float* dr = trt + (size_t)ncol * HW + m0 + 8 * half;
    float* di = tit + (size_t)ncol * HW + m0 + 8 * half;
    *(v8f*)dr = accR;
    *(v8f*)di = accI;
}

// ---------------------------------------------------------------------------
// Stage 2: F = W * T (4 real GEMMs). B tiles (shared by all 4 waves) staged
// in LDS with double-buffered async copies. Power accumulated in registers
// over the image chunk, then symmetric scatter into P with atomics.
// Grid: (17, 8), block 128.
// ---------------------------------------------------------------------------
__global__ void rfl_stage2(const float* __restrict__ Wr,
                           const float* __restrict__ Wi,
                           const float* __restrict__ TrT,
                           const float* __restrict__ TiT,
                           float* __restrict__ P,
                           int nimg) {
    __shared__ float sTr[2][16 * LDSTR];
    __shared__ float sTi[2][16 * LDSTR];

    const int tid  = threadIdx.x;
    const int wave = tid >> 5;
    const int lane = tid & 31;
    const int half = lane >> 4;
    const int r    = lane & 15;
    const int m0 = blockIdx.y * 64 + wave * 16;
    const int n0 = blockIdx.x * 16;

    const int arow = m0 + r;
    const int ncol = n0 + r;

    // cooperative copy mapping: thread -> (column c, 8-float segment part)
    const int cc   = tid >> 3;           // 0..15
    const int part = tid & 7;            // 0..7  (part*8 floats within KC=64)

    const int nchunks = HW / KC;                 // 8
    const int total   = nimg * nchunks;          // flattened (image, chunk)

    float pw[8] = {0.f, 0.f, 0.f, 0.f, 0.f, 0.f, 0.f, 0.f};
    v8f aRR = {}, aII = {}, aRI = {}, aIR = {};

    // issue chunk `it` into buffer it&1
    auto issue = [&](int it) {
        const int i   = it >> 3;                 // image
        const int k0c = (it & 7) * KC;
        const int buf = it & 1;
        const float* gr = TrT + (size_t)i * TROWS * HW
                              + (size_t)(n0 + cc) * HW + k0c + part * 8;
        const float* gi = TiT + (size_t)i * TROWS * HW
                              + (size_t)(n0 + cc) * HW + k0c + part * 8;
        float* dr = &sTr[buf][cc * LDSTR + part * 8];
        float* di = &sTi[buf][cc * LDSTR + part * 8];
        async_b128(dr,     gr);
        async_b128(dr + 4, gr + 4);
        async_b128(di,     gi);
        async_b128(di + 4, gi + 4);
    };

    issue(0);
    for (int it = 0; it < total; ++it) {
        if (it + 1 < total) issue(it + 1);
        wait_async((it + 1 < total) ? 4 : 0);    // chunk `it` complete
        __syncthreads();

        const int buf = it & 1;
        const float* lr = &sTr[buf][r * LDSTR];
        const float* li = &sTi[buf][r * LDSTR];
        #pragma unroll
        for (int s = 0; s < KC / 4; ++s) {
            const int kk = s * 4 + 2 * half;
            v2f wr  = *(const v2f*)(Wr + arow * HW + ((it & 7) * KC + kk));
            v2f wi  = *(const v2f*)(Wi + arow * HW + ((it & 7) * KC + kk));
            v2f btr = *(const v2f*)(lr + kk);
            v2f bti = *(const v2f*)(li + kk);
            aRR = wmma_f32(wr, btr, aRR);
            aII = wmma_f32(wi, bti, aII);
            aRI = wmma_f32(wr, bti, aRI);
            aIR = wmma_f32(wi, btr, aIR);
        }
        if ((it & 7) == 7) {                     // image finished
            #pragma unroll
            for (int v = 0; v < 8; ++v) {
                const float fr = aRR[v] - aII[v];
                const float fi = aRI[v] + aIR[v];
                pw[v] += fr * fr + fi * fi;
            }
            aRR = v8f{}; aII = v8f{}; aRI = v8f{}; aIR = v8f{};
        }
        __syncthreads();                         // protect buffer reuse
    }

    if (ncol <= 256) {
        #pragma unroll
        for (int v = 0; v < 8; ++v) {
            const int row = m0 + v + 8 * half;
            atomicAdd(&P[row * HW + ncol], pw[v]);
            if (ncol >= 1 && ncol <= 255) {
                const int mrow = (HW - row) & (HW - 1);
                atomicAdd(&P[mrow * HW + (HW - ncol)], pw[v]);
            }
        }
    }
}

// ---------------------------------------------------------------------------
// Finalize: loss = sum_b w_b * sum_{u,v} mask[b, fftshift(u,v)] * P[u,v]/NCHW
// ---------------------------------------------------------------------------
__global__ void rfl_finalize(const float* __restrict__ P,
                             const float* __restrict__ masks,
                             const float* __restrict__ wts,
                             float* __restrict__ out) {
    __shared__ float red[1024];
    float w[NB];
    #pragma unroll
    for (int b = 0; b < NB; ++b) w[b] = wts[b];

    float acc = 0.0f;
    for (int idx = threadIdx.x; idx < HW * HW; idx += 1024) {
        const int u  = idx >> 9;
        const int v  = idx & 511;
        const int us = (u + 256) & 511;
        const int vs = (v + 256) & 511;
        const float p = P[idx];
        float wsum = 0.0f;
        #pragma unroll
        for (int b = 0; b < NB; ++b)
            wsum += w[b] * masks[b * HW * HW + us * HW + vs];
        acc += p * wsum;
    }
    red[threadIdx.x] = acc;
    __syncthreads();
    for (int s = 512; s > 0; s >>= 1) {
        if (threadIdx.x < (unsigned)s) red[threadIdx.x] += red[threadIdx.x + s];
        __syncthreads();
    }
    if (threadIdx.x == 0) out[0] = red[0] * (1.0f / (float)NCHW);
}

// ---------------------------------------------------------------------------
extern "C" void kernel_launch(void* const* d_in, const int* in_sizes, int n_in,
                              void* d_out, int out_size, void* d_ws, size_t ws_size,
                              hipStream_t stream) {
    const float* pred  = (const float*)d_in[0];
    const float* gt    = (const float*)d_in[1];
    const float* wts   = (const float*)d_in[2];
    const float* masks = (const float*)d_in[3];

    float* ws  = (float*)d_ws;
    float* Wr  = ws;
    float* Wi  = Wr + HW * HW;
    float* P   = Wi + HW * HW;
    float* TrT = P  + HW * HW;
    float* TiT = TrT + (size_t)CHUNK * TROWS * HW;

    rfl_init<<<(HW * HW) / 256, 256, 0, stream>>>(Wr, Wi, P);

    for (int img0 = 0; img0 < NIMG; img0 += CHUNK) {
        const int ni = (NIMG - img0 < CHUNK) ? (NIMG - img0) : CHUNK;
        dim3 g1(17, 8, ni);
        rfl_stage1<<<g1, 128, 0, stream>>>(pred, gt, Wr, Wi, TrT, TiT, img0);
        dim3 g2(17, 8, 1);
        rfl_stage2<<<g2, 128, 0, stream>>>(Wr, Wi, TrT, TiT, P, ni);
    }

    rfl_finalize<<<1, 1024, 0, stream>>>(P, masks, wts, (float*)d_out);
}